// ResidualGatedGCNModelVRP_23579370455430
// MI455X (gfx1250) — compile-verified
//
#include <hip/hip_runtime.h>

// ---------------------------------------------------------------------------
// Residual Gated GCN (N=1024, D=128, 3 layers) + all-pairs edge classifier.
// GEMMs: CDNA5 V_WMMA_F32_16X16X4_F32 with Tensor-Data-Mover (TDM) LDS
// staging (tensor_load_to_lds + s_wait_tensorcnt), per ISA 7.12 / ch.8.
// ---------------------------------------------------------------------------

typedef float v2f __attribute__((ext_vector_type(2)));
typedef float v8f __attribute__((ext_vector_type(8)));
typedef unsigned int v4u __attribute__((ext_vector_type(4)));
typedef int v4i __attribute__((ext_vector_type(4)));
typedef int v8i __attribute__((ext_vector_type(8)));

#define N_NODES 1024
#define DIM     128
#define KC      32          // K-chunk staged per TDM load
#define LDA_S   34          // A LDS row stride (32 + 2-DWORD TDM pad)
#define LDB_S   48          // B LDS row stride (32 + 16-DWORD TDM pad)

// ---------------------------------------------------------------------------
// TDM 2D tile load: global (row-major, row stride in elements) -> LDS,
// with hardware row padding. Descriptor packing per CDNA5 ISA ch.8:
//   group0: [1:0]=count=1, [63:32]=lds_addr, [120:64]=global_addr, [127:126]=2
//   group1: [17:16]=data_size(4B), [20]=pad_en, [24:22]=pad_interval,
//           [31:25]=pad_amount, [79:48]=tensor_dim0, [111:80]=tensor_dim1,
//           [127:112]=tile_dim0, [143:128]=tile_dim1, [207:160]=dim0_stride
// pad_interval code 4 = every 32 DWORDs; pad_amount code c = (c+1) DWORDs.
// ---------------------------------------------------------------------------
__device__ __forceinline__ void tdm_load_2d(
    const float* lds_dst, const float* gsrc,
    unsigned tile_d0, unsigned tile_d1,
    unsigned tensor_d0, unsigned tensor_d1, unsigned row_stride,
    unsigned pad_amount_code)
{
  unsigned long long ga = (unsigned long long)(uintptr_t)gsrc;
  unsigned lds_addr = (unsigned)(unsigned long long)(uintptr_t)lds_dst;

  v4u g0;
  g0[0] = 1u;                                             // count=1, user mode
  g0[1] = lds_addr;
  g0[2] = (unsigned)ga;
  g0[3] = (unsigned)((ga >> 32) & 0x01FFFFFFull) | (2u << 30);  // type=2

  v8i g1;
  g1[0] = (int)((2u << 16) |            // data_size = 4 bytes
                (1u << 20) |            // pad_enable
                (4u << 22) |            // pad_interval: every 32 DWORDs
                (pad_amount_code << 25));
  g1[1] = (int)((tensor_d0 & 0xFFFFu) << 16);
  g1[2] = (int)((tensor_d0 >> 16) | ((tensor_d1 & 0xFFFFu) << 16));
  g1[3] = (int)((tensor_d1 >> 16) | (tile_d0 << 16));
  g1[4] = (int)(tile_d1 & 0xFFFFu);     // tile_dim2 = 0
  g1[5] = (int)row_stride;              // tensor_dim0_stride (low 32)
  g1[6] = 0;
  g1[7] = 0;

  v4i z4 = {0, 0, 0, 0};
#if __has_include(<hip/amd_detail/amd_gfx1250_TDM.h>)
  v8i z8 = {0, 0, 0, 0, 0, 0, 0, 0};
  __builtin_amdgcn_tensor_load_to_lds(g0, g1, z4, z4, z8, 0);   // 6-arg toolchain
#else
  __builtin_amdgcn_tensor_load_to_lds(g0, g1, z4, z4, 0);       // ROCm 7.2 5-arg
#endif
}

__global__ void zero_loss_kernel(float* __restrict__ p) { *p = 0.0f; }

// h[n][d] = node_features[n][0..2] @ W_embed + b_embed
__global__ __launch_bounds__(128) void embed_kernel(
    const float* __restrict__ nf, const float* __restrict__ W,
    const float* __restrict__ b, float* __restrict__ h)
{
  int n = blockIdx.x, d = threadIdx.x;
  float x0 = nf[n * 3 + 0], x1 = nf[n * 3 + 1], x2 = nf[n * 3 + 2];
  float acc = fmaf(x2, W[2 * DIM + d], b[d]);
  acc = fmaf(x1, W[1 * DIM + d], acc);
  acc = fmaf(x0, W[0 * DIM + d], acc);
  h[(size_t)n * DIM + d] = acc;
}

// ---------------------------------------------------------------------------
// C[M x Nn] = A@B (+ A2@B2) (+ bias). 4 waves/block, 64x32 C tile
// (16x32 per wave, dual accumulator chains). TDM stages A (64xKC, LDS
// stride 34) and B (KCx32, LDS stride 48 -> the two half-wave fragment
// reads land in disjoint bank ranges) per K-chunk; wave 0 issues the TDM
// and waits on TENSORcnt, block barrier publishes the tiles.
// Fragment mapping per ISA 7.12.2 (f32 16x16x4):
//   A: lane m=lane&15, K = 2*(lane>>4)+vgpr ; B: lane n=lane&15, same K
//   C/D: row = q + 8*(lane>>4), col = lane&15
// ---------------------------------------------------------------------------
__global__ __launch_bounds__(128) void wmma_gemm_kernel(
    const float* __restrict__ A, const float* __restrict__ B,
    const float* __restrict__ A2, const float* __restrict__ B2,
    const float* __restrict__ bias, float* __restrict__ C,
    int M, int Nn, int K, int K2)
{
  __shared__ float As[64 * LDA_S];
  __shared__ float Bs[KC * LDB_S];

  const int lane  = threadIdx.x & 31;
  const int wave  = threadIdx.x >> 5;       // 0..3 -> M sub-tile
  const int mr    = lane & 15;
  const int half  = lane >> 4;
  const int tile_n = blockIdx.x * 32;
  const int tile_m = blockIdx.y * 64;

  v8f acc0 = {};
  v8f acc1 = {};

  for (int p = 0; p < 2; ++p) {
    const float* Ap = p ? A2 : A;
    const float* Bp = p ? B2 : B;
    const int    Kp = p ? K2 : K;
    if (Ap == nullptr) continue;

    for (int k0 = 0; k0 < Kp; k0 += KC) {
      if (threadIdx.x < 32) {               // wave 0 drives the TDM
        // A tile: 64 rows x KC, row stride Kp; pad 2 DWORDs -> stride 34
        tdm_load_2d(As, Ap + (size_t)tile_m * Kp + k0,
                    KC, 64, (unsigned)Kp, (unsigned)M, (unsigned)Kp, 1u);
        // B tile: KC rows x 32 cols, row stride Nn; pad 16 DWORDs -> stride 48
        tdm_load_2d(Bs, Bp + (size_t)k0 * Nn + tile_n,
                    32, KC, (unsigned)Nn, (unsigned)Kp, (unsigned)Nn, 15u);
        __builtin_amdgcn_s_wait_tensorcnt(0);
      }
      __syncthreads();                      // tiles visible to all waves

      const float* ar  = &As[(wave * 16 + mr) * LDA_S];
      const float* b0r = &Bs[mr];
      const float* b1r = &Bs[16 + mr];
#pragma unroll
      for (int ks = 0; ks < KC; ks += 4) {
        const int ka = ks + 2 * half;
        v2f a;  a.x  = ar[ka];               a.y  = ar[ka + 1];
        v2f b0; b0.x = b0r[ka * LDB_S];      b0.y = b0r[(ka + 1) * LDB_S];
        v2f b1; b1.x = b1r[ka * LDB_S];      b1.y = b1r[(ka + 1) * LDB_S];
        acc0 = __builtin_amdgcn_wmma_f32_16x16x4_f32(false, a, false, b0, (short)0, acc0, false, false);
        acc1 = __builtin_amdgcn_wmma_f32_16x16x4_f32(false, a, false, b1, (short)0, acc1, false, false);
      }
      __syncthreads();                      // protect LDS before next TDM
    }
  }

  float bias0 = 0.0f, bias1 = 0.0f;
  if (bias != nullptr) {
    bias0 = bias[tile_n + mr];
    bias1 = bias[tile_n + 16 + mr];
  }
#pragma unroll
  for (int q = 0; q < 8; ++q) {
    const int row = tile_m + wave * 16 + q + 8 * half;
    C[(size_t)row * Nn + tile_n + mr]      = acc0[q] + bias0;
    C[(size_t)row * Nn + tile_n + 16 + mr] = acc1[q] + bias1;
  }
}

// h = relu(h + sigmoid(t) * agg)
__global__ __launch_bounds__(256) void gate_update_kernel(
    float* __restrict__ h, const float* __restrict__ t, const float* __restrict__ agg)
{
  const int i = blockIdx.x * 256 + threadIdx.x;
  const float g = 1.0f / (1.0f + __expf(-t[i]));
  h[i] = fmaxf(fmaf(g, agg[i], h[i]), 0.0f);
}

// logits[i][j] = sum_h relu(hi[i][h] + hj[j][h]) * wc2[h] + bc2 (bc1 in hi)
// plus fused squared-error loss accumulation.
__global__ __launch_bounds__(256) void edge_kernel(
    const float* __restrict__ hi, const float* __restrict__ hj,
    const float* __restrict__ adj, const float* __restrict__ ew,
    const float* __restrict__ wc2, const float* __restrict__ bc2,
    float* __restrict__ logits, float* __restrict__ loss)
{
  __shared__ float his[16 * 129];   // +1 pad: column reads hit distinct banks
  __shared__ float hjs[16 * 129];
  __shared__ float wcs[128];
  __shared__ float red[256];

  const int tid = threadIdx.x;
  const int it  = blockIdx.y * 16;
  const int jt  = blockIdx.x * 16;

#pragma unroll
  for (int tl = 0; tl < 8; ++tl) {   // 16 rows x 128 cols, coalesced
    const int e = tid + tl * 256;
    const int r = e >> 7, c = e & 127;
    his[r * 129 + c] = hi[(size_t)(it + r) * DIM + c];
    hjs[r * 129 + c] = hj[(size_t)(jt + r) * DIM + c];
  }
  if (tid < 128) wcs[tid] = wc2[tid];
  __syncthreads();

  const int tx = tid & 15, ty = tid >> 4;
  const float* hr = &his[ty * 129];
  const float* hc = &hjs[tx * 129];
  float acc = 0.0f;
#pragma unroll 16
  for (int k = 0; k < 128; ++k) {
    float v = hr[k] + hc[k];
    v = fmaxf(v, 0.0f);
    acc = fmaf(v, wcs[k], acc);
  }
  const float logit = acc + bc2[0];
  const size_t idx = (size_t)(it + ty) * N_NODES + (jt + tx);
  logits[idx] = logit;

  const float err = logit * adj[idx] - ew[idx];
  red[tid] = err * err;
  __syncthreads();
  for (int s = 128; s > 0; s >>= 1) {
    if (tid < s) red[tid] += red[tid + s];
    __syncthreads();
  }
  if (tid == 0) atomicAdd(loss, red[0]);
}

__global__ void finalize_kernel(const float* __restrict__ loss, float* __restrict__ out)
{
  *out = *loss * (1.0f / ((float)N_NODES * (float)N_NODES));
}

extern "C" void kernel_launch(void* const* d_in, const int* in_sizes, int n_in,
                              void* d_out, int out_size, void* d_ws, size_t ws_size,
                              hipStream_t stream) {
  (void)in_sizes; (void)n_in; (void)out_size; (void)ws_size;

  const float* nf   = (const float*)d_in[0];
  const float* adj  = (const float*)d_in[1];
  const float* ew   = (const float*)d_in[2];
  const float* We   = (const float*)d_in[3];
  const float* be   = (const float*)d_in[4];
  const float* Wu   = (const float*)d_in[5];
  const float* bu   = (const float*)d_in[6];
  const float* Wv   = (const float*)d_in[7];
  const float* bv   = (const float*)d_in[8];
  const float* Wg   = (const float*)d_in[9];
  const float* bg   = (const float*)d_in[10];
  const float* Wc1  = (const float*)d_in[11];
  const float* bc1  = (const float*)d_in[12];
  const float* Wc2  = (const float*)d_in[13];
  const float* bc2  = (const float*)d_in[14];

  float* out = (float*)d_out;        // [1024*1024] logits + [1] loss
  float* ws  = (float*)d_ws;

  const size_t HD = (size_t)N_NODES * DIM;   // 131072 floats
  float* h    = ws;
  float* u    = ws + 1 * HD;
  float* v    = ws + 2 * HD;
  float* agg  = ws + 3 * HD;
  float* t    = ws + 4 * HD;
  float* hi   = ws + 5 * HD;
  float* hj   = ws + 6 * HD;
  float* loss = ws + 7 * HD;

  zero_loss_kernel<<<1, 1, 0, stream>>>(loss);
  embed_kernel<<<N_NODES, DIM, 0, stream>>>(nf, We, be, h);

  const dim3 gGemm(DIM / 32, N_NODES / 64);   // 64x32 C tile per 4-wave block
  for (int l = 0; l < 3; ++l) {
    const float* Wul = Wu + (size_t)l * DIM * DIM;
    const float* Wvl = Wv + (size_t)l * DIM * DIM;
    const float* Wgl = Wg + (size_t)l * 2 * DIM * DIM;
    // u = h @ Wu + bu ; v = h @ Wv + bv
    wmma_gemm_kernel<<<gGemm, 128, 0, stream>>>(h, Wul, nullptr, nullptr, bu + l * DIM, u,
                                                N_NODES, DIM, DIM, 0);
    wmma_gemm_kernel<<<gGemm, 128, 0, stream>>>(h, Wvl, nullptr, nullptr, bv + l * DIM, v,
                                                N_NODES, DIM, DIM, 0);
    // agg = adj @ v   (K = 1024)
    wmma_gemm_kernel<<<gGemm, 128, 0, stream>>>(adj, v, nullptr, nullptr, nullptr, agg,
                                                N_NODES, DIM, N_NODES, 0);
    // t = u @ Wg[:D] + agg @ Wg[D:] + bg   (fused dual GEMM)
    wmma_gemm_kernel<<<gGemm, 128, 0, stream>>>(u, Wgl, agg, Wgl + (size_t)DIM * DIM,
                                                bg + l * DIM, t, N_NODES, DIM, DIM, DIM);
    // h = relu(h + sigmoid(t) * agg)
    gate_update_kernel<<<(N_NODES * DIM) / 256, 256, 0, stream>>>(h, t, agg);
  }

  // hi = h @ Wc1[:D] + bc1 (bias folded) ; hj = h @ Wc1[D:]
  wmma_gemm_kernel<<<gGemm, 128, 0, stream>>>(h, Wc1, nullptr, nullptr, bc1, hi,
                                              N_NODES, DIM, DIM, 0);
  wmma_gemm_kernel<<<gGemm, 128, 0, stream>>>(h, Wc1 + (size_t)DIM * DIM, nullptr, nullptr,
                                              nullptr, hj, N_NODES, DIM, DIM, 0);

  // all-pairs classifier + fused loss
  edge_kernel<<<dim3(N_NODES / 16, N_NODES / 16), 256, 0, stream>>>(
      hi, hj, adj, ew, Wc2, bc2, out, loss);
  finalize_kernel<<<1, 1, 0, stream>>>(loss, out + (size_t)N_NODES * N_NODES);
}